// MultiHashRouter_49855980372026
// MI455X (gfx1250) — compile-verified
//
#include <hip/hip_runtime.h>
#include <hip/hip_bf16.h>

// MultiHashRouter for MI455X (gfx1250, wave32).
// Memory-bound: ~16MB read + ~18MB written => ~1.5us roofline @ 23.3 TB/s.
// Round 4 (final): drop the cross-half shuffle from the sel latch.
//   - guard-free fast path: straight-line 8x unrolled body, loads clusterable
//   - each lane latches its OWN half's routing word at iteration (lane>>1)&7
//     and stores it to a permuted slot inside the wave's 128B sel window
//     (permutation bijects 0..31 -> no shfl_xor(16), no ownHalf select)
//   - wts: one wave-wide 128B store of the constant 0.5
//   - mask row = (0.5,0.5) at aligned pair e0>>1 -> 2 cmps + 2 selects/lane
//   - 8-op quantizer: cvt|abs|, min7, 2x(cmp+cndmask), or, mul24
//
// d_out layout (float32, tuple concatenated):
//   [0,      2N)  selected_experts (N,2)  (ids as float)
//   [2N,     4N)  expert_weights   (N,2)  (all 0.5)
//   [4N, 4N+64N)  expert_masks     (N,64) (0.5 at e0 and e0^1)

#define HIDDEN_DIM 1024
#define N_EXPERTS  64
#define TOKENS_PER_WAVE 16      // 8 iterations x 2 tokens per wave

__device__ __forceinline__ unsigned int weighted_dim(float v, int w) {
    int mag = (int)fabsf(v);          // v_cvt_i32_f32 with |src| modifier (trunc)
    mag = (mag > 7) ? 7 : mag;        // v_min
    int sp = (v > 0.0f) ? 4 : 0;      // v_cmp + v_cndmask (inline 4)
    sp = (v < 0.0f) ? 12 : sp;        // v_cmp + v_cndmask (inline 12)
    // sign(0)=0 handled: v==0 -> mag=0, sp=0 -> dim=0
    return (unsigned int)__mul24(sp | mag, w);   // full-rate 24-bit multiply
}

__device__ __forceinline__ unsigned int hash_pair(
    const float* __restrict__ hs, int token, int b)
{
    const float4 v =
        reinterpret_cast<const float4*>(hs + (size_t)token * HIDDEN_DIM)[b >> 2];
    unsigned int r = weighted_dim(v.x, b + 1)
                   ^ weighted_dim(v.y, b + 2)
                   ^ weighted_dim(v.z, b + 3)
                   ^ weighted_dim(v.w, b + 4);
    // Butterfly XOR across the 16 lanes owning this token (LDS-permute pipe;
    // masks 1,2,4,8 never cross the half-wave boundary).
    r ^= (unsigned int)__shfl_xor((int)r, 1, 32);
    r ^= (unsigned int)__shfl_xor((int)r, 2, 32);
    r ^= (unsigned int)__shfl_xor((int)r, 4, 32);
    r ^= (unsigned int)__shfl_xor((int)r, 8, 32);
    return r;   // routing word, uniform within each 16-lane half
}

__device__ __forceinline__ void store_mask(
    float* __restrict__ mask, int token, int q, int bp, unsigned int r)
{
    const int p = (int)((r & 63u) >> 1);       // both selected experts = pair p
    const float lo = (p == bp)     ? 0.5f : 0.0f;
    const float hi = (p == bp + 1) ? 0.5f : 0.0f;
    float4 m;
    m.x = lo; m.y = lo; m.z = hi; m.w = hi;
    reinterpret_cast<float4*>(mask + (size_t)token * N_EXPERTS)[q] = m;
}

__global__ __launch_bounds__(256) void MultiHashRouter_49855980372026_kernel(
    const float* __restrict__ hs,   // (N, 1024)
    float* __restrict__ sel,        // (N, 2)
    float* __restrict__ wts,        // (N, 2)
    float* __restrict__ mask,       // (N, 64)
    int n_tokens)
{
    const int lane = threadIdx.x & 31;      // wave32
    const int sub  = lane >> 4;             // which token of the pair (0/1)
    const int q    = lane & 15;             // float4 chunk within token
    const int wave = (blockIdx.x * blockDim.x + threadIdx.x) >> 5;
    const int base = wave * TOKENS_PER_WAVE;
    if (base >= n_tokens) return;

    const int b  = q * 4;                   // first dim handled by this lane
    const int bp = q * 2;                   // first expert-pair handled by this lane

    if (base + TOKENS_PER_WAVE <= n_tokens) {
        // ---------- fast path: no per-iteration guards, straight line ----------
        // Lane L latches its OWN half's r at iteration (L>>1)&7 and stores it
        // at slot 2k + (L&1) where token index k = 2*((L>>1)&7) + (L>>4).
        const int itNeed = (lane >> 1) & 7;
        const int selOff = ((lane & 14) << 1) | ((lane >> 3) & 2) | (lane & 1);
        unsigned int keep = 0;

        // gfx1250 prefetch of the wave's second half (pairs 4..7).
        __builtin_prefetch(hs + (size_t)(base +  8 + sub) * HIDDEN_DIM + b, 0, 1);
        __builtin_prefetch(hs + (size_t)(base + 10 + sub) * HIDDEN_DIM + b, 0, 1);
        __builtin_prefetch(hs + (size_t)(base + 12 + sub) * HIDDEN_DIM + b, 0, 1);
        __builtin_prefetch(hs + (size_t)(base + 14 + sub) * HIDDEN_DIM + b, 0, 1);

#pragma unroll
        for (int it = 0; it < TOKENS_PER_WAVE / 2; ++it) {
            const int token = base + it * 2 + sub;
            const unsigned int r = hash_pair(hs, token, b);
            store_mask(mask, token, q, bp, r);
            keep = (itNeed == it) ? r : keep;    // 1 cmp + 1 cndmask
        }

        // One coalesced 128B store each for sel and wts (all 16 tokens).
        const unsigned int e = (keep & 63u) ^ (unsigned int)(lane & 1);
        sel[(size_t)base * 2 + selOff] = (float)e;
        wts[(size_t)base * 2 + lane]   = 0.5f;
    } else {
        // ---------- generic tail path (unused for N=65536) ----------
#pragma unroll 1
        for (int it = 0; it < TOKENS_PER_WAVE / 2; ++it) {
            const int pair0 = base + it * 2;
            if (pair0 >= n_tokens) break;
            int token = pair0 + sub;
            if (token >= n_tokens) token = n_tokens - 1;  // duplicate, same result

            const unsigned int r = hash_pair(hs, token, b);
            store_mask(mask, token, q, bp, r);

            if (q == 0) {
                const int e0 = (int)(r & 63u);
                reinterpret_cast<float2*>(sel)[token] =
                    make_float2((float)e0, (float)(e0 ^ 1));
                reinterpret_cast<float2*>(wts)[token] = make_float2(0.5f, 0.5f);
            }
        }
    }
}

extern "C" void kernel_launch(void* const* d_in, const int* in_sizes, int n_in,
                              void* d_out, int out_size, void* d_ws, size_t ws_size,
                              hipStream_t stream) {
    const float* hs = (const float*)d_in[0];
    const int n_tokens = in_sizes[0] / HIDDEN_DIM;   // 65536

    float* out  = (float*)d_out;
    float* sel  = out;
    float* wts  = out + (size_t)n_tokens * 2;
    float* mask = out + (size_t)n_tokens * 4;

    const int waves   = (n_tokens + TOKENS_PER_WAVE - 1) / TOKENS_PER_WAVE;
    const int threads = waves * 32;
    const int block   = 256;                          // 8 waves per block
    const int grid    = (threads + block - 1) / block;

    MultiHashRouter_49855980372026_kernel<<<grid, block, 0, stream>>>(
        hs, sel, wts, mask, n_tokens);
}